// DCRN_Fusion_30477087932720
// MI455X (gfx1250) — compile-verified
//
#include <hip/hip_runtime.h>
#include <cstdint>

// ---------------------------------------------------------------------------
// out = alpha * segment_sum(adj_val[e] * z_i[adj_col[e]] -> adj_row[e])
//     + (1-alpha) * z_i,           z_i = a*z1 + b*z2,   [N=50000, D=128]
//
// Memory-bound (AI ~1.3 FLOP/B; HBM floor ~6us at 23.3 TB/s). gfx1250 paths:
//   * global_load_async_to_lds_b128 + s_wait_asynccnt (edge staging, ASYNCcnt)
//   * b128 per-lane row traffic (wave32: 1 row = 32 lanes x float4)
//   * hand-emitted global_atomic_add_f32 (no CAS fallback) for the scatter;
//     z_i and out (2 x 25.6MB) stay L2-resident (192MB L2)
//   * non-temporal streaming loads for the read-once z1/z2/a/b
// ---------------------------------------------------------------------------

#define FEAT_D 128
#define TILE_E 256   // edges staged per block (3 KB of LDS)

typedef float v4f __attribute__((ext_vector_type(4)));

// Four hardware f32 atomic-adds to consecutive dwords, one address pair.
__device__ __forceinline__ void gatomic_add4_f32(float* p, v4f m) {
  uint64_t addr = (uint64_t)(uintptr_t)p;
  asm volatile("global_atomic_add_f32 %0, %1, off\n\t"
               "global_atomic_add_f32 %0, %2, off offset:4\n\t"
               "global_atomic_add_f32 %0, %3, off offset:8\n\t"
               "global_atomic_add_f32 %0, %4, off offset:12"
               :: "v"(addr), "v"(m.x), "v"(m.y), "v"(m.z), "v"(m.w)
               : "memory");
}

// Async copy of 16 bytes/lane from global to LDS (tracked by ASYNCcnt).
__device__ __forceinline__ void async_copy_b128(uint32_t lds_off, uint64_t gaddr) {
  asm volatile("global_load_async_to_lds_b128 %0, %1, off"
               :: "v"(lds_off), "v"(gaddr) : "memory");
}
__device__ __forceinline__ void wait_asynccnt0() {
  asm volatile("s_wait_asynccnt 0x0" ::: "memory");
}

// ---------------------------------------------------------------------------
// Kernel 1: z_i = a*z1 + b*z2 (optionally to workspace), out = (1-alpha)*z_i
// ---------------------------------------------------------------------------
template <bool WRITE_ZI>
__global__ void __launch_bounds__(256)
dcrn_elemwise(const v4f* __restrict__ z1, const v4f* __restrict__ z2,
              const v4f* __restrict__ a,  const v4f* __restrict__ b,
              const float* __restrict__ alpha_p,
              v4f* __restrict__ out, v4f* __restrict__ zi, int n4) {
  int i = blockIdx.x * blockDim.x + threadIdx.x;
  if (i >= n4) return;
  float al  = alpha_p[0];
  float om  = 1.0f - al;
  v4f x1, x2, aa, bb;
  if constexpr (WRITE_ZI) {
    // read-once streams: keep them out of L2 (z_i/out need the capacity)
    x1 = __builtin_nontemporal_load(&z1[i]);
    x2 = __builtin_nontemporal_load(&z2[i]);
    aa = __builtin_nontemporal_load(&a[i]);
    bb = __builtin_nontemporal_load(&b[i]);
  } else {
    x1 = z1[i]; x2 = z2[i]; aa = a[i]; bb = b[i];
  }
  v4f z = aa * x1 + bb * x2;                     // v_pk_fma_f32 pairs
  if constexpr (WRITE_ZI) zi[i] = z;             // re-read ~12x by SpMM: RT
  out[i] = om * z;
}

// ---------------------------------------------------------------------------
// Kernel 2: per-edge gather/scale/scatter-add. One block = one TILE_E tile.
// Edge lists staged into LDS via gfx1250 async-to-LDS DMA. One wave per edge
// row: lane L handles floats [4L, 4L+4) of the 128-wide feature row.
// ---------------------------------------------------------------------------
template <bool USE_ZI>
__global__ void __launch_bounds__(256)
dcrn_spmm(const int*   __restrict__ adj_row, const int* __restrict__ adj_col,
          const float* __restrict__ adj_val,
          const float* __restrict__ zi,
          const v4f* __restrict__ z1, const v4f* __restrict__ z2,
          const v4f* __restrict__ a,  const v4f* __restrict__ b,
          const float* __restrict__ alpha_p,
          float* __restrict__ out, int nEdges) {
  __shared__ __align__(16) int   s_row[TILE_E];
  __shared__ __align__(16) int   s_col[TILE_E];
  __shared__ __align__(16) float s_val[TILE_E];

  const int tileBase = blockIdx.x * TILE_E;
  const int rem = (nEdges - tileBase < TILE_E) ? (nEdges - tileBase) : TILE_E;

  if (rem == TILE_E) {
    // Full tile: wave 0 DMAs 3 x 1KB into LDS (2 b128 ops per array).
    if (threadIdx.x < 32) {
      const int l = threadIdx.x;
      const uint32_t lr = (uint32_t)(uintptr_t)(&s_row[0]) + (uint32_t)l * 16u;
      const uint32_t lc = (uint32_t)(uintptr_t)(&s_col[0]) + (uint32_t)l * 16u;
      const uint32_t lv = (uint32_t)(uintptr_t)(&s_val[0]) + (uint32_t)l * 16u;
      const uint64_t gr = (uint64_t)(uintptr_t)(adj_row + tileBase) + (uint64_t)l * 16u;
      const uint64_t gc = (uint64_t)(uintptr_t)(adj_col + tileBase) + (uint64_t)l * 16u;
      const uint64_t gv = (uint64_t)(uintptr_t)(adj_val + tileBase) + (uint64_t)l * 16u;
      async_copy_b128(lr,        gr);
      async_copy_b128(lr + 512u, gr + 512u);
      async_copy_b128(lc,        gc);
      async_copy_b128(lc + 512u, gc + 512u);
      async_copy_b128(lv,        gv);
      async_copy_b128(lv + 512u, gv + 512u);
      wait_asynccnt0();           // per-wave ASYNCcnt; only the issuing wave waits
    }
  } else {
    // Ragged tail tile: cooperative plain fill.
    for (int i = threadIdx.x; i < rem; i += blockDim.x) {
      s_row[i] = adj_row[tileBase + i];
      s_col[i] = adj_col[tileBase + i];
      s_val[i] = adj_val[tileBase + i];
    }
  }
  __syncthreads();

  const float al   = alpha_p[0];
  const int   lane = threadIdx.x & 31;
  const int   wave = threadIdx.x >> 5;

  for (int e = wave; e < rem; e += 8) {           // 8 waves per block
    const int   r = s_row[e];
    const int   c = s_col[e];
    const float w = al * s_val[e];

    v4f v;
    if constexpr (USE_ZI) {
      v = *((const v4f*)(zi + (size_t)c * FEAT_D) + lane);
    } else {
      const size_t o = (size_t)c * (FEAT_D / 4) + (size_t)lane;
      v = a[o] * z1[o] + b[o] * z2[o];
    }

    gatomic_add4_f32(out + (size_t)r * FEAT_D + (size_t)lane * 4u, w * v);
  }
}

// ---------------------------------------------------------------------------
extern "C" void kernel_launch(void* const* d_in, const int* in_sizes, int n_in,
                              void* d_out, int out_size, void* d_ws, size_t ws_size,
                              hipStream_t stream) {
  const float* z1      = (const float*)d_in[0];
  const float* z2      = (const float*)d_in[1];
  const int*   adj_row = (const int*)  d_in[2];
  const int*   adj_col = (const int*)  d_in[3];
  const float* adj_val = (const float*)d_in[4];
  const float* a       = (const float*)d_in[5];
  const float* b       = (const float*)d_in[6];
  const float* alpha   = (const float*)d_in[7];

  const int ND = in_sizes[0];        // N * 128
  const int E  = in_sizes[2];
  const int n4 = ND / 4;
  float* out = (float*)d_out;
  float* zi  = (float*)d_ws;

  const bool use_ws = (ws_size >= (size_t)ND * sizeof(float));

  const int blocks1 = (n4 + 255) / 256;
  if (use_ws) {
    dcrn_elemwise<true><<<blocks1, 256, 0, stream>>>(
        (const v4f*)z1, (const v4f*)z2, (const v4f*)a, (const v4f*)b,
        alpha, (v4f*)out, (v4f*)zi, n4);
  } else {
    dcrn_elemwise<false><<<blocks1, 256, 0, stream>>>(
        (const v4f*)z1, (const v4f*)z2, (const v4f*)a, (const v4f*)b,
        alpha, (v4f*)out, nullptr, n4);
  }

  const int blocks2 = (E + TILE_E - 1) / TILE_E;
  if (use_ws) {
    dcrn_spmm<true><<<blocks2, 256, 0, stream>>>(
        adj_row, adj_col, adj_val, zi,
        nullptr, nullptr, nullptr, nullptr, alpha, out, E);
  } else {
    dcrn_spmm<false><<<blocks2, 256, 0, stream>>>(
        adj_row, adj_col, adj_val, nullptr,
        (const v4f*)z1, (const v4f*)z2, (const v4f*)a, (const v4f*)b,
        alpha, out, E);
  }
}